// LTAE_87694642249753
// MI455X (gfx1250) — compile-verified
//
#include <hip/hip_runtime.h>
#include <hip/hip_bf16.h>

// LTAE encoder for MI455X (gfx1250, wave32).
// GEMMs use V_WMMA_F32_16X16X4_F32 (fp32 in/out — the workload is HBM-bound,
// ~64 GFLOP vs ~0.8 GB traffic, so fp32 tensor ops are free).
// The 491520x256 intermediate is stored bf16 to halve the dominant round-trip;
// BN statistics are taken from fp32 accumulators inside the GEMM (no precision
// loss, and no extra 503MB stats pass).
// b_key is mathematically dead (softmax shift invariance) and is skipped.
// POS_TABLE is a frozen sinusoid; regenerated analytically in-kernel.

typedef float v2f __attribute__((ext_vector_type(2)));
typedef float v8f __attribute__((ext_vector_type(8)));

#define NHEAD 16
#define DK 8
#define DMODEL 256
#define INCH 128
#define TT 60
#define BB 8192
#define MROWS (BB * TT)        // 491520

// ---------------------------------------------------------------- utilities
__global__ void k_zero(float* p, int n) {
    int i = blockIdx.x * blockDim.x + threadIdx.x;
    if (i < n) p[i] = 0.f;
}

// scale/shift from accumulated sum/sumsq (training-mode BN)
__global__ void k_bn_fin(const float* __restrict__ stats,
                         const float* __restrict__ gamma,
                         const float* __restrict__ beta,
                         float* __restrict__ scsh, int C, float invN) {
    int c = blockIdx.x * blockDim.x + threadIdx.x;
    if (c >= C) return;
    float mean = stats[c] * invN;
    float var  = stats[C + c] * invN - mean * mean;
    float sc   = gamma[c] * rsqrtf(var + 1e-5f);
    scsh[c]     = sc;
    scsh[C + c] = beta[c] - mean * sc;
}

// ------------------------------------------------- GEMM1: y1 = x @ w_in.T + b
// M=491520, K=128, N=256. One 16x16 tile per wave, 8 waves (8 m-tiles) per
// block sharing one n-tile so the BN stats reduce to 16 channels per block.
__global__ void k_gemm1(const float* __restrict__ x,
                        const float* __restrict__ w_in,
                        const float* __restrict__ b_in,
                        __hip_bfloat16* __restrict__ y1,
                        float* __restrict__ stats1) {
    __shared__ float s_sum[16], s_ss[16];
    const int tid = threadIdx.x, wv = tid >> 5, ln = tid & 31;
    const int l16 = ln & 15, lh = ln >> 4;
    const int n0 = (blockIdx.x & 15) * 16;
    const int m0 = ((blockIdx.x >> 4) * 8 + wv) * 16;
    if (tid < 16) { s_sum[tid] = 0.f; s_ss[tid] = 0.f; }
    __syncthreads();

    // A lane layout (16x4 f32): m = lane&15, k = 2*(lane>>4)+{0,1}
    // B lane layout (4x16 f32): n = lane&15, k = 2*(lane>>4)+{0,1}
    const float* ap = x    + (size_t)(m0 + l16) * INCH + 2 * lh;
    const float* bp = w_in + (size_t)(n0 + l16) * INCH + 2 * lh;
    v8f c = {};
    for (int k0 = 0; k0 < INCH; k0 += 4) {
        __builtin_prefetch(ap + k0 + 32, 0, 1);
        v2f a = *(const v2f*)(ap + k0);
        v2f b = *(const v2f*)(bp + k0);
        c = __builtin_amdgcn_wmma_f32_16x16x4_f32(false, a, false, b,
                                                  (short)0, c, false, false);
    }
    const float bias = b_in[n0 + l16];
    float s = 0.f, ss = 0.f;
#pragma unroll
    for (int r = 0; r < 8; ++r) {
        float v = c[r] + bias;
        int row = m0 + r + 8 * lh;               // C layout: M = r + 8*(lane>>4)
        y1[(size_t)row * DMODEL + n0 + l16] = __float2bfloat16(v);
        s += v; ss += v * v;
    }
    // combine the two half-wave copies of each channel, then block reduce
    s  += __shfl_xor(s, 16, 32);
    ss += __shfl_xor(ss, 16, 32);
    if (lh == 0) { atomicAdd(&s_sum[l16], s); atomicAdd(&s_ss[l16], ss); }
    __syncthreads();
    if (tid < 16)       atomicAdd(&stats1[n0 + tid], s_sum[tid]);
    else if (tid < 32)  atomicAdd(&stats1[DMODEL + n0 + tid - 16], s_ss[tid - 16]);
}

// --------------------------------- fused BN1 + ReLU + pos + keys + attention
// One block per batch element. h kept transposed [c][t] in LDS so WMMA
// A-fragment reads are lane-contiguous (conflict-free).
__global__ void k_attn(const __hip_bfloat16* __restrict__ y1,
                       const int* __restrict__ positions,
                       const float* __restrict__ scsh1,   // 256 scale, 256 shift
                       const float* __restrict__ w_key,   // (128,256)
                       const float* __restrict__ query,   // (16,8)
                       float* __restrict__ ybuf) {        // (B,256)
    __shared__ float h[DMODEL][TT];   // 61440 B
    __shared__ float att[NHEAD][64];  //  4096 B  (64 KB total)
    const int b = blockIdx.x;
    const int tid = threadIdx.x, wv = tid >> 5, ln = tid & 31;
    const int l16 = ln & 15, lh = ln >> 4;

    const float LN1000_D = 0.43173146f;          // ln(1000)/16
    for (int idx = tid; idx < TT * DMODEL; idx += 256) {
        int t = idx >> 8, c = idx & 255;
        float v = __bfloat162float(y1[((size_t)b * TT + t) * DMODEL + c]);
        v = fmaxf(v * scsh1[c] + scsh1[DMODEL + c], 0.f);     // BN + ReLU
        int p = positions[b * TT + t] + 60;                   // MAX_SHIFT
        int d = c & 15;                                       // tiled 16-dim PE
        float ang = (float)p * __expf(-(float)(d & ~1) * LN1000_D);
        v += (d & 1) ? __cosf(ang) : __sinf(ang);
        h[c][t] = v;
    }
    for (int i = tid; i < NHEAD * 64; i += 256) ((float*)att)[i] = 0.f;
    __syncthreads();

    // k = h @ w_key.T : M=64(60) x K=256 x N=128  -> 32 tiles, 4 per wave.
    // Logits folded immediately: att[head][t] += k * query[head][:] / sqrt(8)
    for (int q = 0; q < 4; ++q) {
        int tile = wv * 4 + q;
        int m0 = (tile >> 3) * 16, n0 = (tile & 7) * 16;
        int row = m0 + l16;                      // t index of this lane's A row
        const float* bp = w_key + (size_t)(n0 + l16) * DMODEL + 2 * lh;
        v8f c = {};
        for (int k0 = 0; k0 < DMODEL; k0 += 4) {
            int k = k0 + 2 * lh;
            v2f a;
            a.x = (row < TT) ? h[k][row]     : 0.f;
            a.y = (row < TT) ? h[k + 1][row] : 0.f;
            v2f bb = *(const v2f*)(bp + k0);
            c = __builtin_amdgcn_wmma_f32_16x16x4_f32(false, a, false, bb,
                                                      (short)0, c, false, false);
        }
        int kc = n0 + l16;                       // 0..127 = head*8 + dk
        float qk = query[kc] * 0.35355339f;      // 1/sqrt(8)
#pragma unroll
        for (int r = 0; r < 8; ++r) {
            int t = m0 + r + 8 * lh;
            if (t < TT) atomicAdd(&att[kc >> 3][t], c[r] * qk);
        }
    }
    __syncthreads();

    // softmax over T per head: 16 threads/head, wave32-native shfl reductions
    {
        int hd = tid >> 4, sl = tid & 15;
        float m = -1e30f;
        for (int t = sl; t < TT; t += 16) m = fmaxf(m, att[hd][t]);
        for (int off = 8; off; off >>= 1) m = fmaxf(m, __shfl_xor(m, off, 16));
        float s = 0.f;
        for (int t = sl; t < TT; t += 16) {
            float e = __expf(att[hd][t] - m);
            att[hd][t] = e; s += e;
        }
        for (int off = 8; off; off >>= 1) s += __shfl_xor(s, off, 16);
        float inv = 1.f / s;
        for (int t = sl; t < TT; t += 16) att[hd][t] *= inv;
    }
    __syncthreads();

    // value pooling: y[c] = sum_t att[c/16][t] * h[t][c]
    float acc = 0.f;
    for (int t = 0; t < TT; ++t) acc += att[tid >> 4][t] * h[tid][t];
    ybuf[(size_t)b * DMODEL + tid] = acc;
}

// ------------------------------------------- GEMM2: z = y @ w_mlp.T + stats2
__global__ void k_gemm2(const float* __restrict__ ybuf,
                        const float* __restrict__ w_mlp,
                        const float* __restrict__ b_mlp,
                        float* __restrict__ z,
                        float* __restrict__ stats2) {
    __shared__ float s_sum[16], s_ss[16];
    const int tid = threadIdx.x, wv = tid >> 5, ln = tid & 31;
    const int l16 = ln & 15, lh = ln >> 4;
    const int n0 = (blockIdx.x & 7) * 16;
    const int m0 = ((blockIdx.x >> 3) * 8 + wv) * 16;
    if (tid < 16) { s_sum[tid] = 0.f; s_ss[tid] = 0.f; }
    __syncthreads();

    const float* ap = ybuf  + (size_t)(m0 + l16) * DMODEL + 2 * lh;
    const float* bp = w_mlp + (size_t)(n0 + l16) * DMODEL + 2 * lh;
    v8f c = {};
    for (int k0 = 0; k0 < DMODEL; k0 += 4) {
        __builtin_prefetch(ap + k0 + 32, 0, 1);
        v2f a = *(const v2f*)(ap + k0);
        v2f b = *(const v2f*)(bp + k0);
        c = __builtin_amdgcn_wmma_f32_16x16x4_f32(false, a, false, b,
                                                  (short)0, c, false, false);
    }
    const float bias = b_mlp[n0 + l16];
    float s = 0.f, ss = 0.f;
#pragma unroll
    for (int r = 0; r < 8; ++r) {
        float v = c[r] + bias;
        int row = m0 + r + 8 * lh;
        z[(size_t)row * 128 + n0 + l16] = v;
        s += v; ss += v * v;
    }
    s  += __shfl_xor(s, 16, 32);
    ss += __shfl_xor(ss, 16, 32);
    if (lh == 0) { atomicAdd(&s_sum[l16], s); atomicAdd(&s_ss[l16], ss); }
    __syncthreads();
    if (tid < 16)      atomicAdd(&stats2[n0 + tid], s_sum[tid]);
    else if (tid < 32) atomicAdd(&stats2[128 + n0 + tid - 16], s_ss[tid - 16]);
}

__global__ void k_out(const float* __restrict__ z,
                      const float* __restrict__ scsh2,
                      float* __restrict__ out) {
    int i = blockIdx.x * blockDim.x + threadIdx.x;
    if (i >= BB * 128) return;
    int c = i & 127;
    out[i] = fmaxf(z[i] * scsh2[c] + scsh2[128 + c], 0.f);
}

// ------------------------------------------------------------------- launch
extern "C" void kernel_launch(void* const* d_in, const int* in_sizes, int n_in,
                              void* d_out, int out_size, void* d_ws, size_t ws_size,
                              hipStream_t stream) {
    const float* x         = (const float*)d_in[0];
    const int*   positions = (const int*)  d_in[1];
    // d_in[2] = gdd (unused by reference path)
    const float* w_in  = (const float*)d_in[3];
    const float* b_in  = (const float*)d_in[4];
    const float* bn1_g = (const float*)d_in[5];
    const float* bn1_b = (const float*)d_in[6];
    const float* w_key = (const float*)d_in[7];
    // d_in[8] = b_key : constant per head across T -> softmax-invariant, skipped
    const float* query = (const float*)d_in[9];
    const float* w_mlp = (const float*)d_in[10];
    const float* b_mlp = (const float*)d_in[11];
    const float* bn2_g = (const float*)d_in[12];
    const float* bn2_b = (const float*)d_in[13];
    float* out = (float*)d_out;

    // workspace carve-up
    char* ws = (char*)d_ws;
    __hip_bfloat16* y1 = (__hip_bfloat16*)ws;          // 491520*256 bf16
    ws += (size_t)MROWS * DMODEL * sizeof(__hip_bfloat16);
    float* stats1 = (float*)ws; ws += 512 * sizeof(float);   // sum|sumsq 256ch
    float* stats2 = (float*)ws; ws += 256 * sizeof(float);   // sum|sumsq 128ch
    float* scsh1  = (float*)ws; ws += 512 * sizeof(float);
    float* scsh2  = (float*)ws; ws += 256 * sizeof(float);
    float* ybuf   = (float*)ws; ws += (size_t)BB * DMODEL * sizeof(float);
    float* z      = (float*)ws; ws += (size_t)BB * 128 * sizeof(float);

    k_zero<<<3, 256, 0, stream>>>(stats1, 768);        // stats1+stats2 contiguous

    k_gemm1<<<(MROWS / 16 / 8) * 16, 256, 0, stream>>>(x, w_in, b_in, y1, stats1);
    k_bn_fin<<<1, 256, 0, stream>>>(stats1, bn1_g, bn1_b, scsh1, DMODEL,
                                    1.0f / (float)MROWS);

    k_attn<<<BB, 256, 0, stream>>>(y1, positions, scsh1, w_key, query, ybuf);

    k_gemm2<<<(BB / 16 / 8) * 8, 256, 0, stream>>>(ybuf, w_mlp, b_mlp, z, stats2);
    k_bn_fin<<<1, 128, 0, stream>>>(stats2, bn2_g, bn2_b, scsh2, 128,
                                    1.0f / (float)BB);

    k_out<<<(BB * 128 + 255) / 256, 256, 0, stream>>>(z, scsh2, out);
}